// HoughVotingLayer_70463233458879
// MI455X (gfx1250) — compile-verified
//
#include <hip/hip_runtime.h>
#include <math.h>

// ---------------------------------------------------------------------------
// Hough voting layer, MI455X (gfx1250, wave32).
// B=16, NHYP=128, H=W=96, NP=9216. Output: (16,2) f32.
// Voting phase expressed as V_WMMA_F32_16X16X4_F32: D = A(128x4) * G(4x9216).
// ---------------------------------------------------------------------------

#define B_    16
#define NHYP  128
#define H_    96
#define W_    96
#define NP    (H_ * W_)          // 9216
#define CHUNKS 16                // pixel chunks per batch
#define TILES_PER_CHUNK 36       // 36 * 16 = 576 pixels per chunk; 16*576 = 9216

typedef float v2f __attribute__((ext_vector_type(2)));
typedef float v8f __attribute__((ext_vector_type(8)));

// ---- block-wide sum over 128 threads (NaN-propagating, as reference) ------
__device__ __forceinline__ float blockSum128(float v, float* red, int n) {
    red[n] = v;
    __syncthreads();
    #pragma unroll
    for (int off = 64; off > 0; off >>= 1) {
        if (n < off) red[n] += red[n + off];
        __syncthreads();
    }
    float r = red[0];
    __syncthreads();
    return r;
}

// ---------------------------------------------------------------------------
// Kernel 1: per-hypothesis intersection Y, mean/std outlier rejection -> Yp,
//           in-mask multiplier, zero vote counters.  grid=(16), block=(128)
// ---------------------------------------------------------------------------
__global__ __launch_bounds__(NHYP)
void hyp_kernel(const float* __restrict__ uv, const int* __restrict__ pidx,
                float* __restrict__ Yp, float* __restrict__ mult,
                unsigned* __restrict__ votes) {
    __shared__ float red[NHYP];
    const int b = blockIdx.x;
    const int n = threadIdx.x;

    const int i0 = pidx[(b * NHYP + n) * 2 + 0];
    const int i1 = pidx[(b * NHYP + n) * 2 + 1];
    const float* uvb = uv + (size_t)b * 2 * NP;

    const float u0 = uvb[i0],      v0 = uvb[NP + i0];
    const float u1 = uvb[i1],      v1 = uvb[NP + i1];
    const float y0 = (float)(i0 / W_), x0 = (float)(i0 % W_);
    const float y1 = (float)(i1 / W_), x1 = (float)(i1 % W_);

    // A = [[u0, -u1], [v0, -v1]],  rhs = (y1-y0, x1-x0);  s = X[0]
    const float a11 = u0, a12 = -u1, a21 = v0, a22 = -v1;
    const float b1 = y1 - y0, b2 = x1 - x0;
    const float det  = a11 * a22 - a12 * a21;
    const float fro2 = a11 * a11 + a12 * a12 + a21 * a21 + a22 * a22;
    float s;
    if (fabsf(det) > 1e-12f * fro2) {
        s = (b1 * a22 - b2 * a12) / det;          // pinv == inverse
    } else if (fro2 > 0.f) {
        s = (a11 * b1 + a21 * b2) / fro2;         // rank-1 pseudoinverse
    } else {
        s = 0.f;                                  // pinv(0) = 0
    }
    const float Yy = y0 + s * u0;
    const float Yx = x0 + s * v0;

    // mean / std (ddof=1) over the 128 hypotheses
    const float my = blockSum128(Yy, red, n) * (1.0f / NHYP);
    const float mx = blockSum128(Yx, red, n) * (1.0f / NHYP);
    const float dy = Yy - my, dx = Yx - mx;
    const float sdy = sqrtf(blockSum128(dy * dy, red, n) * (1.0f / (NHYP - 1)));
    const float sdx = sqrtf(blockSum128(dx * dx, red, n) * (1.0f / (NHYP - 1)));

    // z > 2 on either coordinate (positive side only, matching reference)
    const bool outl = (dy > 2.0f * sdy) || (dx > 2.0f * sdx);
    const float Ypy = outl ? my : Yy;
    const float Ypx = outl ? mx : Yx;

    // trunc(Yp)==pts summed over all cells == 1  <=>  Yp in (-1,96)^2
    const bool im = (Ypy > -1.0f) && (Ypy < (float)H_) &&
                    (Ypx > -1.0f) && (Ypx < (float)W_);

    Yp[(b * NHYP + n) * 2 + 0] = Ypy;
    Yp[(b * NHYP + n) * 2 + 1] = Ypx;
    mult[b * NHYP + n]  = im ? 10.0f : 1.0f;
    votes[b * NHYP + n] = 0u;
}

// ---------------------------------------------------------------------------
// Kernel 2: WMMA voting.  grid=(CHUNKS, B), block=256 (8 waves).
// Wave w owns hypothesis rows [16w, 16w+16); iterates 36 pixel tiles of 16.
//   A (16x4 f32):  lanes 0-15  -> (Yy, Yx) rows,  lanes 16-31 -> (1, 0)
//   B (4x16 f32):  lanes 0-15  -> (u, v),         lanes 16-31 -> (-(py*u+px*v), 0)
//   D (16x16 f32): element [m,n] = vote dot-product; count positives per row.
// Inner loop is straight-line: (py,px) carried incrementally (pixel index
// advances by exactly 16 per tile), B operand built with selects, no divides,
// no EXEC divergence around the WMMA.
// ---------------------------------------------------------------------------
__global__ __launch_bounds__(256)
void vote_kernel(const float* __restrict__ uv, const float* __restrict__ Yp,
                 unsigned* __restrict__ votes) {
    const int b     = blockIdx.y;
    const int chunk = blockIdx.x;
    const int wave  = threadIdx.x >> 5;
    const int lane  = threadIdx.x & 31;
    const int half  = lane >> 4;       // 0: K0/K1 lanes, 1: K2/K3 lanes
    const int lh    = lane & 15;
    const bool khi  = (half != 0);

    // A operand (constant across the whole chunk)
    v2f a;
    if (!khi) {
        const int row = wave * 16 + lh;
        a.x = Yp[(b * NHYP + row) * 2 + 0];     // K=0 : Yy
        a.y = Yp[(b * NHYP + row) * 2 + 1];     // K=1 : Yx
    } else {
        a.x = 1.0f;                             // K=2 : affine 1
        a.y = 0.0f;                             // K=3 : pad
    }

    const float* uvb = uv + (size_t)b * 2 * NP;
    int cnt[8];
    #pragma unroll
    for (int i = 0; i < 8; ++i) cnt[i] = 0;

    const int pchunk = chunk * (TILES_PER_CHUNK * 16);
    const int p0 = pchunk + lh;                 // this lane's first pixel
    float py = (float)(p0 / W_);                // one integer div, outside loop
    float px = (float)(p0 % W_);

    for (int t = 0; t < TILES_PER_CHUNK; ++t) {
        const int p = p0 + t * 16;
        const float u = uvb[p];
        const float v = uvb[NP + p];
        const float cterm = -(py * u + px * v);
        v2f bm;
        bm.x = khi ? cterm : u;                 // K=2 : -c   | K=0 : u
        bm.y = khi ? 0.0f  : v;                 // K=3 : pad  | K=1 : v
        v8f c = {};
        c = __builtin_amdgcn_wmma_f32_16x16x4_f32(
                /*neg_a=*/false, a, /*neg_b=*/false, bm,
                /*c_mod=*/(short)0, c, /*reuse_a=*/false, /*reuse_b=*/false);
        #pragma unroll
        for (int i = 0; i < 8; ++i) cnt[i] += (c[i] > 0.0f) ? 1 : 0;

        // advance this lane's pixel by 16: branchless (py,px) update
        px += 16.0f;
        const bool wrap = (px >= (float)W_);
        px = wrap ? px - (float)W_ : px;
        py = wrap ? py + 1.0f      : py;
    }

    // butterfly-reduce over the 16 lanes sharing each row half
    #pragma unroll
    for (int i = 0; i < 8; ++i) {
        int s = cnt[i];
        s += __shfl_xor(s, 1);
        s += __shfl_xor(s, 2);
        s += __shfl_xor(s, 4);
        s += __shfl_xor(s, 8);
        cnt[i] = s;
    }
    if (lh == 0) {
        #pragma unroll
        for (int i = 0; i < 8; ++i) {
            const int row = wave * 16 + half * 8 + i;
            atomicAdd(&votes[b * NHYP + row], (unsigned)cnt[i]);
        }
    }
}

// ---------------------------------------------------------------------------
// Kernel 3: weights -> normalized weighted mean, flipped to (x, y).
// grid=(16), block=(128)
// ---------------------------------------------------------------------------
__device__ __forceinline__ float nan_to_num(float x) {
    if (x != x) return 0.0f;
    if (x ==  INFINITY) return  __FLT_MAX__;
    if (x == -INFINITY) return -__FLT_MAX__;
    return x;
}

__global__ __launch_bounds__(NHYP)
void finalize_kernel(const float* __restrict__ Yp, const float* __restrict__ mult,
                     const unsigned* __restrict__ votes, float* __restrict__ out) {
    __shared__ float red[NHYP];
    const int b = blockIdx.x;
    const int n = threadIdx.x;

    const float Ypy = Yp[(b * NHYP + n) * 2 + 0];
    const float Ypx = Yp[(b * NHYP + n) * 2 + 1];
    const float w   = mult[b * NHYP + n] * (float)votes[b * NHYP + n];

    const float S = blockSum128(w, red, n);     // normalize BEFORE NaN-zero (as ref)
    float wn = w / S;
    const bool hyp_nan = (Ypy != Ypy) || (Ypx != Ypx);
    if (hyp_nan) wn = 0.0f;

    const float yy = nan_to_num(Ypy);
    const float xx = nan_to_num(Ypx);

    const float sx = blockSum128(xx * wn, red, n);
    const float sy = blockSum128(yy * wn, red, n);
    if (n == 0) {
        out[b * 2 + 0] = sx;   // flipped: x first
        out[b * 2 + 1] = sy;
    }
}

// ---------------------------------------------------------------------------
extern "C" void kernel_launch(void* const* d_in, const int* in_sizes, int n_in,
                              void* d_out, int out_size, void* d_ws, size_t ws_size,
                              hipStream_t stream) {
    const float* uv   = (const float*)d_in[0];   // (16, 2, 96, 96) f32
    /* d_in[1]: mask (bool) — unused by the reference math */
    const int*   pidx = (const int*)d_in[2];     // (16, 128, 2) i32
    float* out = (float*)d_out;                  // (16, 2) f32

    float*    Yp    = (float*)d_ws;                      // 16*128*2 f32
    float*    mult  = Yp + (size_t)B_ * NHYP * 2;        // 16*128   f32
    unsigned* votes = (unsigned*)(mult + (size_t)B_ * NHYP); // 16*128 u32

    hyp_kernel<<<B_, NHYP, 0, stream>>>(uv, pidx, Yp, mult, votes);
    vote_kernel<<<dim3(CHUNKS, B_), 256, 0, stream>>>(uv, Yp, votes);
    finalize_kernel<<<B_, NHYP, 0, stream>>>(Yp, mult, votes, out);
}